// SHDSpikingNetwork_51908974739500
// MI455X (gfx1250) — compile-verified
//
#include <hip/hip_runtime.h>

typedef __attribute__((ext_vector_type(2))) float v2f;
typedef __attribute__((ext_vector_type(8))) float v8f;
typedef __attribute__((ext_vector_type(4))) unsigned int u32x4;
typedef __attribute__((ext_vector_type(4))) int i32x4;
typedef __attribute__((ext_vector_type(8))) int i32x8;

#define B_    256
#define T_    100
#define I_    700
#define H_    1024
#define O_    20
#define OPAD  32

#define ALPHA_C 0.8187307530779818f   // exp(-0.001/0.005)
#define BETA_C  0.9048374180359595f   // exp(-0.001/0.01)

#define BKC  32                 // K chunk
#define LDT  36                 // padded LDS stride: 32 DWORDs + 4 DWORD pad (TDM pad_interval=32, pad_amount=4)
#define TILE (64 * LDT)         // one LDS tile buffer (floats)

#if __has_builtin(__builtin_amdgcn_tensor_load_to_lds)
#define USE_TDM 1
#else
#define USE_TDM 0
#endif

// Low 32 bits of a generic pointer to __shared__ == LDS byte offset (ISA 10.2 aperture rule).
__device__ __forceinline__ unsigned int lds_off(const float* p) {
    return (unsigned int)(unsigned long long)(uintptr_t)p;
}

#if USE_TDM
// Issue one TDM 2D tile load: 64 rows x 32 DWORDs of row-major A (stride lda floats)
// into LDS rows padded to LDT floats. kvalid = K - kb: TDM returns zeros beyond
// tensor_dim0, giving ragged-K zero padding for free. (D# packing per ISA ch.8.)
__device__ __forceinline__ void tdm_issue_A(const float* gptr, unsigned int lds_byte,
                                            int kvalid, long lda)
{
    unsigned long long ga = (unsigned long long)(uintptr_t)gptr;
    u32x4 g0;
    g0.x = 1u;                                                // count=1, user mode
    g0.y = lds_byte;                                          // lds_addr [63:32]
    g0.z = (unsigned int)ga;                                  // global_addr[31:0]
    g0.w = (unsigned int)((ga >> 32) & 0x1FFFFFFull) | (2u << 30); // addr[56:32] | type=2

    unsigned int td0 = (unsigned int)kvalid;                  // tensor_dim0 (K left)
    unsigned int td1 = 64u;                                   // tensor_dim1 (rows)
    unsigned long long s0 = (unsigned long long)lda;          // tensor_dim0_stride
    i32x8 g1;
    // data_size=2 (4B) | pad_enable | pad_interval=4 (32 DW) | pad_amount=3 (4 DW)
    g1[0] = (int)((2u << 16) | (1u << 20) | (4u << 22) | (3u << 25));
    g1[1] = (int)((td0 & 0xFFFFu) << 16);                     // tensor_dim0 lo16 at [63:48]
    g1[2] = (int)(((td0 >> 16) & 0xFFFFu) | ((td1 & 0xFFFFu) << 16));
    g1[3] = (int)(((td1 >> 16) & 0xFFFFu) | (32u << 16));     // tile_dim0=32 at [127:112]
    g1[4] = (int)(64u);                                       // tile_dim1=64; tile_dim2=0
    g1[5] = (int)(unsigned int)(s0 & 0xFFFFFFFFull);          // stride0 [191:160]
    g1[6] = (int)(unsigned int)((s0 >> 32) & 0xFFFFull);      // stride0 hi; stride1 lo = 0
    g1[7] = 0;
    i32x4 z4 = {0, 0, 0, 0};
#if defined(__clang_major__) && (__clang_major__ >= 23)
    i32x8 z8 = {0, 0, 0, 0, 0, 0, 0, 0};
    __builtin_amdgcn_tensor_load_to_lds(g0, g1, z4, z4, z8, 0);
#else
    __builtin_amdgcn_tensor_load_to_lds(g0, g1, z4, z4, 0);
#endif
}
#endif

// Stage A chunk: TDM (wave 0 issues, DMA direct to LDS) or thread fallback.
__device__ __forceinline__ void stage_A(const float* __restrict__ A, long lda,
                                        int kb, int K, float* dstA,
                                        int tid, int wave)
{
#if USE_TDM
    (void)tid;
    if (wave == 0) tdm_issue_A(A + kb, lds_off(dstA), K - kb, lda);
#else
    (void)wave;
#pragma unroll
    for (int i = 0; i < 16; ++i) {
        int e = tid + i * 128, row = e >> 5, kk = e & 31, k = kb + kk;
        dstA[row * LDT + kk] = (k < K) ? A[(long)row * lda + k] : 0.f;
    }
#endif
}

// ---------------------------------------------------------------------------
// 64x64 fp32 tile GEMM accumulator: V_WMMA_F32_16X16X4_F32, double-buffered
// LDS, TDM staging for A, register-prefetched (float4) transposed staging for B.
// Block = 128 threads (4 waves); wave w computes rows [16w,16w+16) x 64 cols.
// ---------------------------------------------------------------------------
__device__ __forceinline__ void gemm64x64_acc(
    const float* __restrict__ A, long lda,
    const float* __restrict__ Bm, long ldb,
    int K, float* sA /*2*TILE*/, float* sB /*2*TILE*/, v8f acc[4])
{
    const int tid  = threadIdx.x;        // 0..127
    const int wave = tid >> 5;
    const int lane = tid & 31;
    const int half = lane >> 4;
    const int ln16 = lane & 15;
    const int mrow = wave * 16 + ln16;
    const int NC   = (K + BKC - 1) / BKC;

    __syncthreads();                     // defensive: LDS reuse across calls

    float4 breg[4];
    // ---- prologue: chunk 0 into buffer 0 ----
    stage_A(A, lda, 0, K, sA, tid, wave);
#pragma unroll
    for (int i = 0; i < 4; ++i) {        // B: 32 k-rows x 64 n, float4 per thread-slot
        int e = tid + i * 128, kr = e >> 4, n = (e & 15) * 4;
        breg[i] = (kr < K) ? *(const float4*)&Bm[(long)kr * ldb + n]
                           : make_float4(0.f, 0.f, 0.f, 0.f);
    }
#pragma unroll
    for (int i = 0; i < 4; ++i) {        // transpose into sB[n][k]
        int e = tid + i * 128, kr = e >> 4, n = (e & 15) * 4;
        sB[(n + 0) * LDT + kr] = breg[i].x;
        sB[(n + 1) * LDT + kr] = breg[i].y;
        sB[(n + 2) * LDT + kr] = breg[i].z;
        sB[(n + 3) * LDT + kr] = breg[i].w;
    }
#if USE_TDM
    __builtin_amdgcn_s_wait_tensorcnt(0);
#endif
    __syncthreads();

    for (int c = 0; c < NC; ++c) {
        const int cur = c & 1;
        float* cA = sA + cur * TILE;
        float* cB = sB + cur * TILE;
        float* pA = sA + (1 - cur) * TILE;
        float* pB = sB + (1 - cur) * TILE;
        const bool hasNext = (c + 1 < NC);

        if (hasNext) {                   // prefetch chunk c+1 (overlaps WMMA below)
            int kb = (c + 1) * BKC;
            stage_A(A, lda, kb, K, pA, tid, wave);
#pragma unroll
            for (int i = 0; i < 4; ++i) {
                int e = tid + i * 128, kr = e >> 4, n = (e & 15) * 4;
                int k = kb + kr;
                breg[i] = (k < K) ? *(const float4*)&Bm[(long)k * ldb + n]
                                  : make_float4(0.f, 0.f, 0.f, 0.f);
            }
        }
        // ---- compute on current buffers ----
#pragma unroll
        for (int ks = 0; ks < BKC; ks += 4) {
            v2f af = *(const v2f*)&cA[mrow * LDT + ks + 2 * half];
#pragma unroll
            for (int j = 0; j < 4; ++j) {
                v2f bf = *(const v2f*)&cB[(j * 16 + ln16) * LDT + ks + 2 * half];
                acc[j] = __builtin_amdgcn_wmma_f32_16x16x4_f32(
                    false, af, false, bf, (short)0, acc[j], false, false);
            }
        }
        if (hasNext) {
#pragma unroll
            for (int i = 0; i < 4; ++i) {
                int e = tid + i * 128, kr = e >> 4, n = (e & 15) * 4;
                pB[(n + 0) * LDT + kr] = breg[i].x;
                pB[(n + 1) * LDT + kr] = breg[i].y;
                pB[(n + 2) * LDT + kr] = breg[i].z;
                pB[(n + 3) * LDT + kr] = breg[i].w;
            }
#if USE_TDM
            __builtin_amdgcn_s_wait_tensorcnt(0);
#endif
            __syncthreads();
        }
    }
}

// ---------------------------------------------------------------------------
// Phase 1: h1pre[b*T+t][h] = inputs[b][t][:] @ w1   (M=25600, K=700, N=1024)
// ---------------------------------------------------------------------------
__global__ void __launch_bounds__(128) gemm_input_kernel(
    const float* __restrict__ A, const float* __restrict__ W1,
    float* __restrict__ C)
{
    __shared__ __align__(16) float sA[2 * TILE];
    __shared__ __align__(16) float sB[2 * TILE];
    v8f acc[4] = {};
    const long row0 = (long)blockIdx.y * 64;
    const long col0 = (long)blockIdx.x * 64;
    gemm64x64_acc(A + row0 * I_, I_, W1 + col0, H_, I_, sA, sB, acc);

    const int wave = threadIdx.x >> 5;
    const int lane = threadIdx.x & 31;
    const int half = lane >> 4;
    const int ln16 = lane & 15;
#pragma unroll
    for (int j = 0; j < 4; ++j)
#pragma unroll
        for (int r = 0; r < 8; ++r)
            C[(row0 + wave * 16 + r + 8 * half) * H_ + col0 + j * 16 + ln16] = acc[j][r];
}

// ---------------------------------------------------------------------------
// Phase 2: one LIF time step. Grid covers B x H (64x64 tiles).
// FUSE=true folds inputs[:,t,:] @ w1 in (no h1pre workspace). Double-buffered
// spike matrices (outRead/outWrite) avoid the intra-launch RAW race.
// ---------------------------------------------------------------------------
template <bool FUSE>
__global__ void __launch_bounds__(128) lif_step_kernel(
    const float* __restrict__ inputs, const float* __restrict__ w1,
    const float* __restrict__ h1pre,  const float* __restrict__ v1,
    const float* __restrict__ outRead, float* __restrict__ outWrite,
    float* __restrict__ syn, float* __restrict__ mem,
    float* __restrict__ mem_rec, float* __restrict__ spk_rec, int t)
{
    __shared__ __align__(16) float sA[2 * TILE];
    __shared__ __align__(16) float sB[2 * TILE];
    v8f acc[4] = {};
    const long b0 = (long)blockIdx.y * 64;   // batch tile
    const long h0 = (long)blockIdx.x * 64;   // hidden tile

    if (FUSE)   // inputs[b][t][:] @ w1 : A row stride between batches = T*I
        gemm64x64_acc(inputs + b0 * (long)T_ * I_ + (long)t * I_, (long)T_ * I_,
                      w1 + h0, H_, I_, sA, sB, acc);
    // recurrent drive: out_prev @ v1
    gemm64x64_acc(outRead + b0 * H_, H_, v1 + h0, H_, H_, sA, sB, acc);

    const int wave = threadIdx.x >> 5;
    const int lane = threadIdx.x & 31;
    const int half = lane >> 4;
    const int ln16 = lane & 15;
#pragma unroll
    for (int j = 0; j < 4; ++j)
#pragma unroll
        for (int r = 0; r < 8; ++r) {
            long b = b0 + wave * 16 + r + 8 * half;
            long h = h0 + j * 16 + ln16;
            long rec_i = (b * T_ + t) * (long)H_ + h;
            long st_i  = b * H_ + h;
            float h1 = acc[j][r] + (FUSE ? 0.f : h1pre[rec_i]);
            float m  = mem[st_i];
            float s  = syn[st_i];
            float spk = (m - 1.f) > 0.f ? 1.f : 0.f;  // spike from mem BEFORE update
            mem_rec[rec_i]  = m;                      // record pre-update mem
            spk_rec[rec_i]  = spk;
            outWrite[st_i]  = spk;
            syn[st_i]       = ALPHA_C * s + h1;
            mem[st_i]       = (BETA_C * m + s) * (1.f - spk);
        }
}

// ---------------------------------------------------------------------------
// Phase 3a: h2[b*T+t][0..32) = spk_rec[b][t][:] @ w2 (O=20 padded to 32).
// ---------------------------------------------------------------------------
__global__ void __launch_bounds__(32) readout_gemm_kernel(
    const float* __restrict__ spk, const float* __restrict__ w2,
    float* __restrict__ h2)
{
    const int lane = threadIdx.x;
    const int half = lane >> 4;
    const int ln16 = lane & 15;
    const long row0 = (long)blockIdx.x * 16;
    const float* Ab = spk + row0 * H_;
    v8f acc0 = {}, acc1 = {};
    const int n1 = 16 + ln16;

    for (int k = 0; k < H_; k += 4) {
        v2f af = *(const v2f*)&Ab[(long)ln16 * H_ + k + 2 * half];
        int krow = k + 2 * half;
        v2f bf0;
        bf0.x = w2[(long)krow * O_ + ln16];
        bf0.y = w2[(long)(krow + 1) * O_ + ln16];
        v2f bf1 = {0.f, 0.f};
        if (n1 < O_) {
            bf1.x = w2[(long)krow * O_ + n1];
            bf1.y = w2[(long)(krow + 1) * O_ + n1];
        }
        acc0 = __builtin_amdgcn_wmma_f32_16x16x4_f32(false, af, false, bf0, (short)0, acc0, false, false);
        acc1 = __builtin_amdgcn_wmma_f32_16x16x4_f32(false, af, false, bf1, (short)0, acc1, false, false);
    }
#pragma unroll
    for (int r = 0; r < 8; ++r) {
        long row = row0 + r + 8 * half;
        h2[row * OPAD + ln16]      = acc0[r];
        h2[row * OPAD + 16 + ln16] = acc1[r];
    }
}

// ---------------------------------------------------------------------------
// Phase 3b: leaky readout scan; thread per (b, o). out_rec is [B, T+1, O].
// ---------------------------------------------------------------------------
__global__ void readout_scan_kernel(const float* __restrict__ h2,
                                    float* __restrict__ out_rec)
{
    int idx = blockIdx.x * blockDim.x + threadIdx.x;
    if (idx >= B_ * O_) return;
    int b = idx / O_, o = idx % O_;
    float flt = 0.f, out = 0.f;
    out_rec[(long)b * (T_ + 1) * O_ + o] = 0.f;   // prepended zero step
    for (int t = 0; t < T_; ++t) {
        float h  = h2[((long)b * T_ + t) * OPAD + o];
        float nf = ALPHA_C * flt + h;
        float no = BETA_C * out + flt;            // uses OLD flt (ref semantics)
        out_rec[(long)b * (T_ + 1) * O_ + (t + 1) * O_ + o] = no;
        flt = nf; out = no;
    }
}

// ---------------------------------------------------------------------------
extern "C" void kernel_launch(void* const* d_in, const int* in_sizes, int n_in,
                              void* d_out, int out_size, void* d_ws, size_t ws_size,
                              hipStream_t stream) {
    (void)in_sizes; (void)n_in; (void)out_size;
    const float* inputs = (const float*)d_in[0];
    const float* w1     = (const float*)d_in[1];
    const float* w2     = (const float*)d_in[2];
    const float* v1     = (const float*)d_in[3];

    float* out_rec = (float*)d_out;                       // [256,101,20]
    float* mem_rec = out_rec + (long)B_ * (T_ + 1) * O_;  // [256,100,1024]
    float* spk_rec = mem_rec + (long)B_ * T_ * H_;        // [256,100,1024]

    const long nBH    = (long)B_ * H_;                    // 262144
    const long nH1    = (long)B_ * T_ * H_;               // 26214400
    const long nH2    = (long)B_ * T_ * OPAD;             // 819200
    const size_t needBig = (size_t)(nH1 + 4 * nBH + nH2) * sizeof(float);
    const bool fuse = (ws_size < needBig);                // fallback: no h1pre buffer

    float* w = (float*)d_ws;
    float* h1pre = nullptr;
    if (!fuse) { h1pre = w; w += nH1; }
    float* syn  = w;
    float* mem  = w + nBH;
    float* outA = w + 2 * nBH;
    float* outB = w + 3 * nBH;
    float* h2   = w + 4 * nBH;

    hipMemsetAsync(syn,  0, nBH * sizeof(float), stream);
    hipMemsetAsync(mem,  0, nBH * sizeof(float), stream);
    hipMemsetAsync(outA, 0, nBH * sizeof(float), stream);

    if (!fuse) {
        dim3 g(H_ / 64, (B_ * T_) / 64);                  // 16 x 400
        gemm_input_kernel<<<g, 128, 0, stream>>>(inputs, w1, h1pre);
    }

    dim3 gs(H_ / 64, B_ / 64);                            // 16 x 4
    for (int t = 0; t < T_; ++t) {
        const float* oR = (t & 1) ? outB : outA;
        float*       oW = (t & 1) ? outA : outB;
        if (fuse)
            lif_step_kernel<true><<<gs, 128, 0, stream>>>(
                inputs, w1, nullptr, v1, oR, oW, syn, mem, mem_rec, spk_rec, t);
        else
            lif_step_kernel<false><<<gs, 128, 0, stream>>>(
                nullptr, nullptr, h1pre, v1, oR, oW, syn, mem, mem_rec, spk_rec, t);
    }

    readout_gemm_kernel<<<(B_ * T_) / 16, 32, 0, stream>>>(spk_rec, w2, h2);
    readout_scan_kernel<<<(B_ * O_ + 255) / 256, 256, 0, stream>>>(h2, out_rec);
}